// GCNEncoder_42640435314981
// MI455X (gfx1250) — compile-verified
//
#include <hip/hip_runtime.h>

typedef float v8f __attribute__((ext_vector_type(8)));
typedef float v2f __attribute__((ext_vector_type(2)));

#define N_CH 128

// ---------------------------------------------------------------------------
// GEMM: H[M, 128] = X[M, 128] @ W[128, 128], full fp32 via V_WMMA_F32_16X16X4_F32
// One block = 256 threads = 8 wave32; block handles a 16-row tile of X staged
// in LDS; wave w computes the 16x16 output tile at columns [16w, 16w+16).
// ---------------------------------------------------------------------------
__global__ __launch_bounds__(256) void gemm_wmma_f32(
    const float* __restrict__ X, const float* __restrict__ W,
    float* __restrict__ H, int nRows)
{
  __shared__ float xs[16][N_CH + 4];   // +4 pad: A-frag column reads conflict-free

  const int rowBase = blockIdx.x * 16;
  const int tid = threadIdx.x;

  // Cooperative load of the 16x128 X tile (512 float4, 2 per thread).
  for (int i = tid; i < 16 * (N_CH / 4); i += 256) {
    const int r  = i / (N_CH / 4);
    const int c4 = (i % (N_CH / 4)) * 4;
    float4 v = make_float4(0.f, 0.f, 0.f, 0.f);
    const int gr = rowBase + r;
    if (gr < nRows) v = *(const float4*)(X + (size_t)gr * N_CH + c4);
    xs[r][c4 + 0] = v.x; xs[r][c4 + 1] = v.y;
    xs[r][c4 + 2] = v.z; xs[r][c4 + 3] = v.w;
  }
  __syncthreads();

  const int wave  = tid >> 5;          // 0..7 -> output column tile
  const int lane  = tid & 31;
  const int n0    = wave * 16;
  const int lm    = lane & 15;         // M index (A) / N index (B) inside tile
  const int khalf = (lane >> 4) * 2;   // lanes 0-15: K=0,1 ; lanes 16-31: K=2,3

  v8f acc = {0.f, 0.f, 0.f, 0.f, 0.f, 0.f, 0.f, 0.f};

  for (int k0 = 0; k0 < N_CH; k0 += 4) {
    // A fragment: 16x4 fp32, one row per lane-mod-16, K split across lane halves.
    v2f a;
    a.x = xs[lm][k0 + khalf];
    a.y = xs[lm][k0 + khalf + 1];
    // B fragment: 4x16 fp32, row-major W[k][n]; N per lane, K split across halves.
    const float* wp = W + (size_t)(k0 + khalf) * N_CH + (n0 + lm);
    v2f b;
    b.x = wp[0];
    b.y = wp[N_CH];
    // (neg_a, A, neg_b, B, c_mod, C, reuse_a, reuse_b)
    acc = __builtin_amdgcn_wmma_f32_16x16x4_f32(
        false, a, false, b, (short)0, acc, false, false);
  }

  // D layout: VGPR r -> lanes 0-15 hold M=r, lanes 16-31 hold M=r+8; N = n0+lm.
  const int mBase = (lane >> 4) * 8;
#pragma unroll
  for (int r = 0; r < 8; ++r) {
    const int gr = rowBase + mBase + r;
    if (gr < nRows) H[(size_t)gr * N_CH + n0 + lm] = acc[r];
  }
}

// ---------------------------------------------------------------------------
// Degree / normalization kernels
// ---------------------------------------------------------------------------
__global__ void deg_init(float* __restrict__ deg, int n) {
  const int i = blockIdx.x * blockDim.x + threadIdx.x;
  if (i < n) deg[i] = 1.0f;                      // self-loop
}

__global__ void deg_count(const int* __restrict__ dst, float* __restrict__ deg, int e) {
  const int i = blockIdx.x * blockDim.x + threadIdx.x;
  if (i < e) atomicAdd(&deg[dst[i]], 1.0f);
}

__global__ void deg_rsqrt(float* __restrict__ deg, int n) {
  const int i = blockIdx.x * blockDim.x + threadIdx.x;
  if (i < n) deg[i] = rsqrtf(deg[i]);            // deg >= 1 always
}

// ---------------------------------------------------------------------------
// Aggregation: out[d] = sum_{e: dst=d} h[src_e] * dinv[src]*dinv[dst]
//              + h[d] * dinv[d]^2            (self loop)
// One thread per (edge-or-node, 4-channel group); 32 lanes cover a 128-ch row.
// ---------------------------------------------------------------------------
__global__ void self_loop_init(const float* __restrict__ H,
                               const float* __restrict__ dinv,
                               float* __restrict__ out, int outStride, int n)
{
  const int idx  = blockIdx.x * blockDim.x + threadIdx.x;
  const int node = idx >> 5;
  const int c4   = (idx & 31) * 4;
  if (node >= n) return;
  float s = dinv[node];
  s = s * s;
  const float4 v = *(const float4*)(H + (size_t)node * N_CH + c4);
  float* o = out + (size_t)node * outStride + c4;
  o[0] = v.x * s; o[1] = v.y * s; o[2] = v.z * s; o[3] = v.w * s;   // plain store: clears poison
}

__global__ void edge_agg(const float* __restrict__ H,
                         const float* __restrict__ dinv,
                         const int* __restrict__ src,
                         const int* __restrict__ dst,
                         float* __restrict__ out, int outStride, int nEdges)
{
  const int idx = blockIdx.x * blockDim.x + threadIdx.x;
  const int e   = idx >> 5;
  const int c4  = (idx & 31) * 4;
  if (e >= nEdges) return;
  const int s = src[e];
  const int d = dst[e];
  const float coef = dinv[s] * dinv[d];
  const float4 v = *(const float4*)(H + (size_t)s * N_CH + c4);   // L2-resident gather
  float* o = out + (size_t)d * outStride + c4;
  atomicAdd(o + 0, v.x * coef);
  atomicAdd(o + 1, v.y * coef);
  atomicAdd(o + 2, v.z * coef);
  atomicAdd(o + 3, v.w * coef);
}

// Layer-1 epilogue: h1 = relu(agg + b1), kept contiguous in ws AND copied to d_out[:,0:128].
__global__ void bias_relu_copy(float* __restrict__ agg,
                               const float* __restrict__ bias,
                               float* __restrict__ outLeft,  // d_out base, stride 2*N_CH
                               int n)
{
  const int idx  = blockIdx.x * blockDim.x + threadIdx.x;
  const int node = idx >> 5;
  const int c4   = (idx & 31) * 4;
  if (node >= n) return;
  float4 v = *(const float4*)(agg + (size_t)node * N_CH + c4);
  v.x = fmaxf(v.x + bias[c4 + 0], 0.f);
  v.y = fmaxf(v.y + bias[c4 + 1], 0.f);
  v.z = fmaxf(v.z + bias[c4 + 2], 0.f);
  v.w = fmaxf(v.w + bias[c4 + 3], 0.f);
  *(float4*)(agg + (size_t)node * N_CH + c4) = v;                 // h1 for layer-2 GEMM
  *(float4*)(outLeft + (size_t)node * (2 * N_CH) + c4) = v;       // d_out[:, 0:128]
}

// Layer-2 epilogue: in place on d_out[:, 128:256].
__global__ void bias_relu_inplace(float* __restrict__ outRight,  // d_out + 128, stride 2*N_CH
                                  const float* __restrict__ bias,
                                  int n)
{
  const int idx  = blockIdx.x * blockDim.x + threadIdx.x;
  const int node = idx >> 5;
  const int c4   = (idx & 31) * 4;
  if (node >= n) return;
  float* p = outRight + (size_t)node * (2 * N_CH) + c4;
  float4 v = *(const float4*)p;
  v.x = fmaxf(v.x + bias[c4 + 0], 0.f);
  v.y = fmaxf(v.y + bias[c4 + 1], 0.f);
  v.z = fmaxf(v.z + bias[c4 + 2], 0.f);
  v.w = fmaxf(v.w + bias[c4 + 3], 0.f);
  *(float4*)p = v;
}

// ---------------------------------------------------------------------------
static inline int cdiv(long long a, long long b) { return (int)((a + b - 1) / b); }

extern "C" void kernel_launch(void* const* d_in, const int* in_sizes, int n_in,
                              void* d_out, int out_size, void* d_ws, size_t ws_size,
                              hipStream_t stream) {
  const float* x  = (const float*)d_in[0];
  const float* W1 = (const float*)d_in[1];
  const float* b1 = (const float*)d_in[2];
  const float* W2 = (const float*)d_in[3];
  const float* b2 = (const float*)d_in[4];
  const int* edges = (const int*)d_in[5];

  const int n = in_sizes[0] / N_CH;      // 100000
  const int e = in_sizes[5] / 2;         // 1600000
  const int* src = edges;
  const int* dst = edges + e;
  float* out = (float*)d_out;

  // Workspace layout (256B aligned): dinv[n] | bufH[n*128] | bufA[n*128]
  char* wsb = (char*)d_ws;
  size_t off = 0;
  float* dinv = (float*)(wsb + off);
  off += ((size_t)n * 4 + 255) & ~(size_t)255;
  float* bufH = (float*)(wsb + off);
  off += ((size_t)n * N_CH * 4 + 255) & ~(size_t)255;
  float* bufA = (float*)(wsb + off);

  const int TB = 256;
  const long long nodeWork = (long long)n * 32;  // (node, ch4) pairs
  const long long edgeWork = (long long)e * 32;  // (edge, ch4) pairs

  // --- symmetric normalization coefficients ---
  deg_init<<<cdiv(n, TB), TB, 0, stream>>>(dinv, n);
  deg_count<<<cdiv(e, TB), TB, 0, stream>>>(dst, dinv, e);
  deg_rsqrt<<<cdiv(n, TB), TB, 0, stream>>>(dinv, n);

  // --- layer 1 ---
  gemm_wmma_f32<<<cdiv(n, 16), TB, 0, stream>>>(x, W1, bufH, n);
  self_loop_init<<<cdiv(nodeWork, TB), TB, 0, stream>>>(bufH, dinv, bufA, N_CH, n);
  edge_agg<<<cdiv(edgeWork, TB), TB, 0, stream>>>(bufH, dinv, src, dst, bufA, N_CH, e);
  bias_relu_copy<<<cdiv(nodeWork, TB), TB, 0, stream>>>(bufA, b1, out, n);

  // --- layer 2 (aggregate straight into d_out[:, 128:256], stride 256) ---
  gemm_wmma_f32<<<cdiv(n, 16), TB, 0, stream>>>(bufA, W2, bufH, n);
  self_loop_init<<<cdiv(nodeWork, TB), TB, 0, stream>>>(bufH, dinv, out + N_CH, 2 * N_CH, n);
  edge_agg<<<cdiv(edgeWork, TB), TB, 0, stream>>>(bufH, dinv, src, dst, out + N_CH, 2 * N_CH, e);
  bias_relu_inplace<<<cdiv(nodeWork, TB), TB, 0, stream>>>(out + N_CH, b2, n);
}